// InvariantPointAttention_79491254714789
// MI455X (gfx1250) — compile-verified
//
#include <hip/hip_runtime.h>
#include <math.h>

// ---------------- problem constants ----------------
#define NN   768
#define CSd  384
#define CZd  128
#define CHd  16
#define Hd   12
#define PQd  4
#define PVd  8
#define DCAT 2112   // H*(CH + PV*4 + CZ) = 12*(16+32+128)

typedef __attribute__((ext_vector_type(2))) float v2f;
typedef __attribute__((ext_vector_type(8))) float v8f;

// ---------------- workspace layout (floats) ----------------
constexpr size_t OFF_Q     = 0;                                  // 768*192
constexpr size_t OFF_KV    = OFF_Q    + (size_t)NN * 192;        // 768*384
constexpr size_t OFF_QP    = OFF_KV   + (size_t)NN * 384;        // 768*144
constexpr size_t OFF_KVP   = OFF_QP   + (size_t)NN * 144;        // 768*432
constexpr size_t OFF_QA    = OFF_KVP  + (size_t)NN * 432;        // 12*768*32
constexpr size_t OFF_KA    = OFF_QA   + (size_t)Hd * NN * 32;
constexpr size_t OFF_ROWT  = OFF_KA   + (size_t)Hd * NN * 32;    // 12*768
constexpr size_t OFF_COLT  = OFF_ROWT + (size_t)Hd * NN;
constexpr size_t OFF_VBT   = OFF_COLT + (size_t)Hd * NN;         // 12*48*768 (col-major per head)
constexpr size_t OFF_A     = OFF_VBT  + (size_t)Hd * 48 * NN;    // 12*768*768 (bias->logits->softmax)
constexpr size_t OFF_OACC  = OFF_A    + (size_t)Hd * NN * NN;    // 768*12*48
constexpr size_t OFF_OPAIR = OFF_OACC + (size_t)NN * Hd * 48;    // 768*12*128
constexpr size_t OFF_CAT   = OFF_OPAIR+ (size_t)NN * Hd * CZd;   // 768*2112
constexpr size_t OFF_WBP   = OFF_CAT  + (size_t)NN * DCAT;       // 16*128 zero-padded w_b
constexpr size_t OFF_ZPAD  = OFF_WBP  + 16 * 128;                // 1024-float zero row

// ---------------- WMMA f32 16x16x4 tile loops ----------------
// Computes D += A(16xK) * B(Kx16) with B[k][m] = W[m][k] (W row-major, M x K).
// Fragment layout (ISA 7.12.2, 32-bit):
//   A: lane l / l+16 holds row M=l, regs = K = 2*half + {0,1}
//   B: lane l / l+16 holds col N=l, regs = K = 2*half + {0,1}
//   C/D: reg r, lane l(+16) -> row r + 8*half, col l
// Out-of-range rows are redirected (once, outside the loop) to a zeroed row.
__device__ __forceinline__ v8f wmma_loop_contig(const float* __restrict__ A, int lda, int nvalid,
                                                const float* __restrict__ W, int ldw, int m0,
                                                int mvalid, const float* __restrict__ zpad,
                                                int K) {
  const int lane = threadIdx.x & 31;
  const int l    = lane & 15;
  const int half = lane >> 4;
  const float* ap = (l < nvalid) ? (A + (size_t)l * lda) : zpad;
  const float* wp = ((m0 + l) < mvalid) ? (W + (size_t)(m0 + l) * ldw) : zpad;
  ap += 2 * half;
  wp += 2 * half;
  v8f acc = {};
#pragma unroll 4
  for (int k0 = 0; k0 < K; k0 += 4) {
    v2f a = *(const v2f*)(ap + k0);
    v2f b = *(const v2f*)(wp + k0);
    acc = __builtin_amdgcn_wmma_f32_16x16x4_f32(false, a, false, b, (short)0, acc,
                                                false, false);
  }
  return acc;
}

// Same, but B element W[k][m] lives at Wcol[m + k*kstride] (k-strided, m contiguous, always valid).
__device__ __forceinline__ v8f wmma_loop_bstride(const float* __restrict__ A, size_t lda,
                                                 int nvalid,
                                                 const float* __restrict__ Wcol, int kstride,
                                                 int m0, const float* __restrict__ zpad,
                                                 int K) {
  const int lane = threadIdx.x & 31;
  const int l    = lane & 15;
  const int half = lane >> 4;
  const float* ap = (l < nvalid) ? (A + (size_t)l * lda) : zpad;
  ap += 2 * half;
  const float* wp = Wcol + (m0 + l);
  v8f acc = {};
#pragma unroll 4
  for (int k0 = 0; k0 < K; k0 += 4) {
    v2f a = *(const v2f*)(ap + k0);
    v2f b;
    b.x = wp[(size_t)(k0 + 2 * half)     * kstride];
    b.y = wp[(size_t)(k0 + 2 * half + 1) * kstride];
    acc = __builtin_amdgcn_wmma_f32_16x16x4_f32(false, a, false, b, (short)0, acc,
                                                false, false);
  }
  return acc;
}

// ---------------- init: zero pad row + zero-padded w_b (16 x 128) ----------------
__global__ void k_init(const float* __restrict__ w_b, float* __restrict__ wbp,
                       float* __restrict__ zpad) {
  int t = blockIdx.x * blockDim.x + threadIdx.x;
  if (t < 16 * CZd) wbp[t] = (t < Hd * CZd) ? w_b[t] : 0.0f;
  if (t < 1024) zpad[t] = 0.0f;
}

// ---------------- generic GEMM with bias: C[n,m] = sum_k A[n,k]*W[m,k] + bias[m] ----------------
// All call sites have M (cols) a multiple of 16 and full row tiles -> no guards needed.
__global__ void k_gemm_bias(const float* __restrict__ A, int lda,
                            const float* __restrict__ W, int ldw,
                            const float* __restrict__ bias,
                            float* __restrict__ C, int ldc, int K) {
  const int m0 = blockIdx.x * 16;
  const int n0 = blockIdx.y * 16;
  v8f acc = wmma_loop_contig(A + (size_t)n0 * lda, lda, 16, W, ldw, m0, 1 << 30, nullptr, K);
  const int lane = threadIdx.x & 31, l = lane & 15, half = lane >> 4;
  const int m = m0 + l;
  const float bv = bias[m];
#pragma unroll
  for (int r = 0; r < 8; ++r) {
    int n = n0 + r + 8 * half;
    C[(size_t)n * ldc + m] = acc[r] + bv;
  }
}

// ---------------- pair bias: aws[h,i,j] = sum_c z[i,j,c]*wbp[h,c] + b_b[h] ----------------
__global__ void k_zbias(const float* __restrict__ z, const float* __restrict__ wbp,
                        const float* __restrict__ b_b, float* __restrict__ aws) {
  const int j0 = blockIdx.x * 16;
  const int i  = blockIdx.y;
  v8f acc = wmma_loop_contig(z + ((size_t)i * NN + j0) * CZd, CZd, 16,
                             wbp, CZd, 0, 1 << 30, nullptr, CZd);
  const int lane = threadIdx.x & 31, l = lane & 15, half = lane >> 4;
  const int h = l;
  if (h < Hd) {
    float bb = b_b[h];
#pragma unroll
    for (int r = 0; r < 8; ++r) {
      int j = j0 + r + 8 * half;
      aws[((size_t)h * NN + i) * NN + j] = acc[r] + bb;
    }
  }
}

// ---------------- point prep: rotate points, build augmented Q/K, row/col terms, VBT ----------------
__global__ void k_prep(const float* __restrict__ qw, const float* __restrict__ kvw,
                       const float* __restrict__ qpw, const float* __restrict__ kvpw,
                       const float* __restrict__ rot, const float* __restrict__ trans,
                       const float* __restrict__ head_w,
                       float* __restrict__ QA, float* __restrict__ KA,
                       float* __restrict__ rowt, float* __restrict__ colt,
                       float* __restrict__ VBT) {
  int tid = blockIdx.x * blockDim.x + threadIdx.x;
  if (tid >= NN * Hd) return;
  const int i = tid / Hd;
  const int h = tid % Hd;
  float R[9];
#pragma unroll
  for (int r = 0; r < 9; ++r) R[r] = rot[i * 9 + r];
  const float t0 = trans[i * 3 + 0], t1 = trans[i * 3 + 1], t2 = trans[i * 3 + 2];
  // hw = softplus(head_weights) * sqrt(1/(3 * PQ * 9/2)) = softplus * sqrt(1/54)
  const float hw = logf(1.0f + expf(head_w[h])) * 0.13608276348795434f;
  const float s1 = 0.14433756729740643f;  // sqrt(1/(3*CH)) = sqrt(1/48)

  // --- QA: [ q*s1 (16) | hw*q_pts (12) | 0 (4) ]
  float* qa = QA + ((size_t)h * NN + i) * 32;
#pragma unroll
  for (int c = 0; c < CHd; ++c) qa[c] = qw[(size_t)i * (Hd * CHd) + h * CHd + c] * s1;
  float q2 = 0.0f;
#pragma unroll
  for (int p = 0; p < PQd; ++p) {
    const float* v = qpw + (size_t)i * (Hd * PQd * 3) + (h * PQd + p) * 3;
    float x = v[0], y = v[1], zz = v[2];
    float px = R[0] * x + R[1] * y + R[2] * zz + t0;
    float py = R[3] * x + R[4] * y + R[5] * zz + t1;
    float pz = R[6] * x + R[7] * y + R[8] * zz + t2;
    qa[16 + p * 3 + 0] = hw * px;
    qa[16 + p * 3 + 1] = hw * py;
    qa[16 + p * 3 + 2] = hw * pz;
    q2 += px * px + py * py + pz * pz;
  }
  qa[28] = qa[29] = qa[30] = qa[31] = 0.0f;
  rowt[h * NN + i] = -0.5f * hw * q2;

  // --- KA: [ k (16) | k_pts (12) | 0 (4) ],   VBT cols: [ v (16) | v_pts (24) | 0 (8) ]
  float* ka = KA + ((size_t)h * NN + i) * 32;
  float* vbt = VBT + (size_t)h * 48 * NN;  // vbt[c*NN + j]
#pragma unroll
  for (int c = 0; c < CHd; ++c) {
    ka[c] = kvw[(size_t)i * (Hd * 2 * CHd) + h * 2 * CHd + c];
    vbt[(size_t)c * NN + i] = kvw[(size_t)i * (Hd * 2 * CHd) + h * 2 * CHd + CHd + c];
  }
  float k2 = 0.0f;
#pragma unroll
  for (int p = 0; p < PQd + PVd; ++p) {
    const float* v = kvpw + (size_t)i * (Hd * (PQd + PVd) * 3) + (h * (PQd + PVd) + p) * 3;
    float x = v[0], y = v[1], zz = v[2];
    float px = R[0] * x + R[1] * y + R[2] * zz + t0;
    float py = R[3] * x + R[4] * y + R[5] * zz + t1;
    float pz = R[6] * x + R[7] * y + R[8] * zz + t2;
    if (p < PQd) {
      ka[16 + p * 3 + 0] = px;
      ka[16 + p * 3 + 1] = py;
      ka[16 + p * 3 + 2] = pz;
      k2 += px * px + py * py + pz * pz;
    } else {
      int pv = p - PQd;
      vbt[(size_t)(16 + pv * 3 + 0) * NN + i] = px;
      vbt[(size_t)(16 + pv * 3 + 1) * NN + i] = py;
      vbt[(size_t)(16 + pv * 3 + 2) * NN + i] = pz;
    }
  }
  ka[28] = ka[29] = ka[30] = ka[31] = 0.0f;
  colt[h * NN + i] = -0.5f * hw * k2;
#pragma unroll
  for (int c = 40; c < 48; ++c) vbt[(size_t)c * NN + i] = 0.0f;
}

// ---------------- logits: aws <- QA.KA^T + sqrt(1/3)*aws + rowt + colt + mask ----------------
__global__ void k_logits(const float* __restrict__ QA, const float* __restrict__ KA,
                         const float* __restrict__ rowt, const float* __restrict__ colt,
                         const float* __restrict__ mask, float* __restrict__ aws) {
  const int j0 = blockIdx.x * 16;
  const int i0 = blockIdx.y * 16;
  const int h  = blockIdx.z;
  v8f acc = wmma_loop_contig(QA + ((size_t)h * NN + i0) * 32, 32, 16,
                             KA + (size_t)h * NN * 32, 32, j0, 1 << 30, nullptr, 32);
  const int lane = threadIdx.x & 31, l = lane & 15, half = lane >> 4;
  const int j = j0 + l;
  const float mj = mask[j];
  const float ct = colt[h * NN + j];
  const float s3 = 0.5773502691896258f;  // sqrt(1/3)
#pragma unroll
  for (int r = 0; r < 8; ++r) {
    int ii = i0 + r + 8 * half;
    size_t idx = ((size_t)h * NN + ii) * NN + j;
    float v = acc[r] + s3 * aws[idx] + rowt[h * NN + ii] + ct +
              100000.0f * (mask[ii] * mj - 1.0f);
    aws[idx] = v;
  }
}

// ---------------- row softmax over j (row length 768) ----------------
__global__ void k_softmax(float* __restrict__ aws) {
  __shared__ float red[256];
  float* row = aws + (size_t)blockIdx.x * NN;
  const int t = threadIdx.x;
  float v0 = row[t], v1 = row[t + 256], v2 = row[t + 512];
  float m = fmaxf(v0, fmaxf(v1, v2));
  red[t] = m; __syncthreads();
  for (int s = 128; s > 0; s >>= 1) { if (t < s) red[t] = fmaxf(red[t], red[t + s]); __syncthreads(); }
  m = red[0]; __syncthreads();
  v0 = expf(v0 - m); v1 = expf(v1 - m); v2 = expf(v2 - m);
  red[t] = v0 + v1 + v2; __syncthreads();
  for (int s = 128; s > 0; s >>= 1) { if (t < s) red[t] += red[t + s]; __syncthreads(); }
  float inv = 1.0f / red[0];
  row[t] = v0 * inv; row[t + 256] = v1 * inv; row[t + 512] = v2 * inv;
}

// ---------------- oacc[i,h,c48] = sum_j a[h,i,j] * VBT[h][c][j] ----------------
__global__ void k_attn_v(const float* __restrict__ aws, const float* __restrict__ VBT,
                         float* __restrict__ oacc) {
  const int c0 = blockIdx.x * 16;
  const int i0 = blockIdx.y * 16;
  const int h  = blockIdx.z;
  v8f acc = wmma_loop_contig(aws + ((size_t)h * NN + i0) * NN, NN, 16,
                             VBT + (size_t)h * 48 * NN, NN, c0, 1 << 30, nullptr, NN);
  const int lane = threadIdx.x & 31, l = lane & 15, half = lane >> 4;
  const int c = c0 + l;
#pragma unroll
  for (int r = 0; r < 8; ++r) {
    int ii = i0 + r + 8 * half;
    oacc[((size_t)ii * Hd + h) * 48 + c] = acc[r];
  }
}

// ---------------- opair[i,h,c] = sum_j a[h,i,j] * z[i,j,c] ----------------
__global__ void k_attn_z(const float* __restrict__ aws, const float* __restrict__ z,
                         const float* __restrict__ zpad, float* __restrict__ opair) {
  const int c0 = blockIdx.x * 16;
  const int i  = blockIdx.y;
  // A rows = h (stride NN*NN, only 12 valid -> zpad), cols = j ;
  // B[k=j][m=c] = z[i*NN*CZ + j*CZ + c]
  v8f acc = wmma_loop_bstride(aws + (size_t)i * NN, (size_t)NN * NN, Hd,
                              z + (size_t)i * NN * CZd, CZd, c0, zpad, NN);
  const int lane = threadIdx.x & 31, l = lane & 15, half = lane >> 4;
  const int c = c0 + l;
#pragma unroll
  for (int r = 0; r < 8; ++r) {
    int h = r + 8 * half;
    if (h < Hd) opair[((size_t)i * Hd + h) * CZd + c] = acc[r];
  }
}

// ---------------- finalize: inverse rotation, norms, concat ----------------
__global__ void k_finalize(const float* __restrict__ oacc, const float* __restrict__ opair,
                           const float* __restrict__ rot, const float* __restrict__ trans,
                           float* __restrict__ cat) {
  int tid = blockIdx.x * blockDim.x + threadIdx.x;
  if (tid >= NN * Hd) return;
  const int i = tid / Hd;
  const int h = tid % Hd;
  float R[9];
#pragma unroll
  for (int r = 0; r < 9; ++r) R[r] = rot[i * 9 + r];
  const float t0 = trans[i * 3 + 0], t1 = trans[i * 3 + 1], t2 = trans[i * 3 + 2];
  float* crow = cat + (size_t)i * DCAT;
  const float* oa = oacc + ((size_t)i * Hd + h) * 48;
#pragma unroll
  for (int c = 0; c < CHd; ++c) crow[h * CHd + c] = oa[c];
#pragma unroll
  for (int p = 0; p < PVd; ++p) {
    float gx = oa[16 + p * 3 + 0] - t0;
    float gy = oa[16 + p * 3 + 1] - t1;
    float gz = oa[16 + p * 3 + 2] - t2;
    // local = R^T * g
    float lx = R[0] * gx + R[3] * gy + R[6] * gz;
    float ly = R[1] * gx + R[4] * gy + R[7] * gz;
    float lz = R[2] * gx + R[5] * gy + R[8] * gz;
    crow[192 + (h * PVd + p) * 3 + 0] = lx;
    crow[192 + (h * PVd + p) * 3 + 1] = ly;
    crow[192 + (h * PVd + p) * 3 + 2] = lz;
    crow[480 + h * PVd + p] = sqrtf(lx * lx + ly * ly + lz * lz + 1e-8f);
  }
  const float* op = opair + ((size_t)i * Hd + h) * CZd;
  for (int c = 0; c < CZd; ++c) crow[576 + h * CZd + c] = op[c];
}

// ---------------- host launcher ----------------
extern "C" void kernel_launch(void* const* d_in, const int* in_sizes, int n_in,
                              void* d_out, int out_size, void* d_ws, size_t ws_size,
                              hipStream_t stream) {
  const float* s       = (const float*)d_in[0];
  const float* z       = (const float*)d_in[1];
  const float* rot     = (const float*)d_in[2];
  const float* trans   = (const float*)d_in[3];
  const float* mask    = (const float*)d_in[4];
  const float* w_q     = (const float*)d_in[5];
  const float* b_q     = (const float*)d_in[6];
  const float* w_kv    = (const float*)d_in[7];
  const float* b_kv    = (const float*)d_in[8];
  const float* w_qp    = (const float*)d_in[9];
  const float* b_qp    = (const float*)d_in[10];
  const float* w_kvp   = (const float*)d_in[11];
  const float* b_kvp   = (const float*)d_in[12];
  const float* w_b     = (const float*)d_in[13];
  const float* b_b     = (const float*)d_in[14];
  const float* head_w  = (const float*)d_in[15];
  const float* w_out   = (const float*)d_in[16];
  const float* b_out   = (const float*)d_in[17];

  float* ws = (float*)d_ws;
  float* q_ws   = ws + OFF_Q;
  float* kv_ws  = ws + OFF_KV;
  float* qp_ws  = ws + OFF_QP;
  float* kvp_ws = ws + OFF_KVP;
  float* QA     = ws + OFF_QA;
  float* KA     = ws + OFF_KA;
  float* rowt   = ws + OFF_ROWT;
  float* colt   = ws + OFF_COLT;
  float* VBT    = ws + OFF_VBT;
  float* aws    = ws + OFF_A;
  float* oacc   = ws + OFF_OACC;
  float* opair  = ws + OFF_OPAIR;
  float* cat    = ws + OFF_CAT;
  float* wbp    = ws + OFF_WBP;
  float* zpad   = ws + OFF_ZPAD;

  const int NT = NN / 16;  // 48 row tiles

  // 0) zero pad-row + zero-padded w_b
  k_init<<<17, 128, 0, stream>>>(w_b, wbp, zpad);

  // 1) projections from s
  k_gemm_bias<<<dim3(12, NT), 32, 0, stream>>>(s, CSd, w_q,   CSd, b_q,   q_ws,   192, CSd);
  k_gemm_bias<<<dim3(24, NT), 32, 0, stream>>>(s, CSd, w_kv,  CSd, b_kv,  kv_ws,  384, CSd);
  k_gemm_bias<<<dim3( 9, NT), 32, 0, stream>>>(s, CSd, w_qp,  CSd, b_qp,  qp_ws,  144, CSd);
  k_gemm_bias<<<dim3(27, NT), 32, 0, stream>>>(s, CSd, w_kvp, CSd, b_kvp, kvp_ws, 432, CSd);

  // 2) point transforms + augmented Q/K + VBT
  k_prep<<<(NN * Hd + 127) / 128, 128, 0, stream>>>(q_ws, kv_ws, qp_ws, kvp_ws, rot, trans,
                                                    head_w, QA, KA, rowt, colt, VBT);

  // 3) pair bias from z (streams z once)
  k_zbias<<<dim3(NT, NN), 32, 0, stream>>>(z, wbp, b_b, aws);

  // 4) attention logits (QK + point term via augmented dot) in place
  k_logits<<<dim3(NT, NT, Hd), 32, 0, stream>>>(QA, KA, rowt, colt, mask, aws);

  // 5) softmax over j
  k_softmax<<<Hd * NN, 256, 0, stream>>>(aws);

  // 6) attention * [v | v_pts]
  k_attn_v<<<dim3(3, NT, Hd), 32, 0, stream>>>(aws, VBT, oacc);

  // 7) attention * z (streams z a second time)
  k_attn_z<<<dim3(CZd / 16, NN), 32, 0, stream>>>(aws, z, zpad, opair);

  // 8) inverse rotation, norms, concat
  k_finalize<<<(NN * Hd + 127) / 128, 128, 0, stream>>>(oacc, opair, rot, trans, cat);

  // 9) output projection
  k_gemm_bias<<<dim3(24, NT), 32, 0, stream>>>(cat, DCAT, w_out, DCAT, b_out,
                                               (float*)d_out, CSd, DCAT);
}